// HGTConvNet_75333726371855
// MI455X (gfx1250) — compile-verified
//
#include <hip/hip_runtime.h>

#define N_NODES 100000
#define D 128
#define H 4
#define DH 32
#define E_EDGES 600000
#define BATCH 50000
#define INV_SQRT_DH 0.17677669529663687f /* 1/sqrt(32) */

typedef __attribute__((ext_vector_type(2))) float v2f;
typedef __attribute__((ext_vector_type(4))) float v4f;
typedef __attribute__((ext_vector_type(8))) float v8f;

// ---------------------------------------------------------------------------
// Generic 128-wide GEMM: Y[M,128] = X[M,128] @ W[128,128] + bias, with
// optional fused sigmoid-skip residual epilogue (mode 1).
// One wave computes a 16x128 strip via V_WMMA_F32_16X16X4_F32.
// The wave's entire A strip (32 floats/lane) is preloaded into registers as
// 16 b64 loads so the WMMA stream never stalls on global latency; B fragments
// come from the LDS-staged weight matrix.
// ---------------------------------------------------------------------------
__global__ __launch_bounds__(256) void gemm128_kernel(
    const float* __restrict__ X, const float* __restrict__ W,
    const float* __restrict__ bias, float* __restrict__ Y,
    int M, int mode, const float* __restrict__ skipPtr,
    const float* __restrict__ xres)
{
    __shared__ float sW[D * D];
    {   // cooperative stage of W into LDS (64 KB)
        const v4f* src = (const v4f*)W;
        v4f* dst = (v4f*)sW;
        for (int i = threadIdx.x; i < D * D / 4; i += 256) dst[i] = src[i];
    }
    __syncthreads();

    const int wave = threadIdx.x >> 5;
    const int lane = threadIdx.x & 31;
    const int m0   = (blockIdx.x * 8 + wave) * 16;
    if (m0 >= M) return;                    // wave-uniform: EXEC stays all-1s
    const int l  = lane & 15;
    const int hi = lane >> 4;

    // A strip preload: lane covers row (m0+l), K pairs (4t+2hi, 4t+2hi+1)
    const float* xr = X + (size_t)(m0 + l) * D + 2 * hi;
    v2f aReg[16];
#pragma unroll
    for (int t = 0; t < 16; ++t) {
        aReg[t].x = xr[4 * t];
        aReg[t].y = xr[4 * t + 1];
    }

    v8f acc[8] = {};
#pragma unroll
    for (int t = 0; t < 16; ++t) {
        const int k0 = 4 * t;
#pragma unroll
        for (int n = 0; n < 8; ++n) {
            v2f b;
            b.x = sW[(k0 + 2 * hi) * D + n * 16 + l];
            b.y = sW[(k0 + 2 * hi + 1) * D + n * 16 + l];
            acc[n] = __builtin_amdgcn_wmma_f32_16x16x4_f32(
                false, aReg[t], false, b, (short)0, acc[n], false, false);
        }
    }

    float aSkip = 0.0f, bSkip = 0.0f;
    if (mode == 1) {
        aSkip = 1.0f / (1.0f + __expf(-skipPtr[0]));
        bSkip = 1.0f - aSkip;
    }
#pragma unroll
    for (int n = 0; n < 8; ++n) {
        const int col = n * 16 + l;
        const float bi = bias[col];
#pragma unroll
        for (int j = 0; j < 8; ++j) {
            const int row = m0 + j + 8 * hi;
            float y = acc[n][j] + bi;
            if (mode == 1)
                y = aSkip * y + bSkip * xres[(size_t)row * D + col];
            Y[(size_t)row * D + col] = y;
        }
    }
}

// ---------------------------------------------------------------------------
// Per-head relation transform: Y[n,h,:] = X[n,h,:] @ R[h]  (R: [H,32,32])
// One wave per (16-row tile, head): 16x32 output, K=32, A preloaded (8 b64).
// ---------------------------------------------------------------------------
__global__ __launch_bounds__(256) void relgemm_kernel(
    const float* __restrict__ X, const float* __restrict__ R,
    float* __restrict__ Y, int M)
{
    const int wave = threadIdx.x >> 5;
    const int lane = threadIdx.x & 31;
    const int w    = blockIdx.x * 8 + wave;
    const int m0   = (w >> 2) * 16;
    const int h    = w & 3;
    if (m0 >= M) return;                    // wave-uniform
    const int l  = lane & 15;
    const int hi = lane >> 4;

    const float* xr = X + (size_t)(m0 + l) * D + h * DH + 2 * hi;
    const float* Rh = R + h * DH * DH;

    v2f aReg[8];
#pragma unroll
    for (int t = 0; t < 8; ++t) {
        aReg[t].x = xr[4 * t];
        aReg[t].y = xr[4 * t + 1];
    }

    v8f acc0 = {}, acc1 = {};
#pragma unroll
    for (int t = 0; t < 8; ++t) {
        const int k0 = 4 * t;
        v2f b0, b1;
        b0.x = Rh[(k0 + 2 * hi) * DH + l];
        b0.y = Rh[(k0 + 2 * hi + 1) * DH + l];
        b1.x = Rh[(k0 + 2 * hi) * DH + 16 + l];
        b1.y = Rh[(k0 + 2 * hi + 1) * DH + 16 + l];
        acc0 = __builtin_amdgcn_wmma_f32_16x16x4_f32(false, aReg[t], false, b0,
                                                     (short)0, acc0, false, false);
        acc1 = __builtin_amdgcn_wmma_f32_16x16x4_f32(false, aReg[t], false, b1,
                                                     (short)0, acc1, false, false);
    }
#pragma unroll
    for (int j = 0; j < 8; ++j) {
        const int row = m0 + j + 8 * hi;
        Y[(size_t)row * D + h * DH + l]      = acc0[j];
        Y[(size_t)row * D + h * DH + 16 + l] = acc1[j];
    }
}

// ---------------------------------------------------------------------------
// Segment softmax over edges (order-preserving int encoding for float max)
// ---------------------------------------------------------------------------
__device__ __forceinline__ int ordf(float f) {
    int i = __float_as_int(f);
    return (i >= 0) ? i : (i ^ 0x7fffffff);
}
__device__ __forceinline__ float unordf(int i) {
    return __int_as_float((i >= 0) ? i : (i ^ 0x7fffffff));
}

__global__ void init_seg_kernel(int* __restrict__ m, float* __restrict__ s, int n) {
    int gid = blockIdx.x * blockDim.x + threadIdx.x;
    if (gid < n) { m[gid] = (int)0x80000000; s[gid] = 0.0f; }
}

__global__ void zero_kernel(float* __restrict__ p, int n) {
    int gid = blockIdx.x * blockDim.x + threadIdx.x;
    if (gid < n) p[gid] = 0.0f;
}

__global__ void edge_logits_kernel(const int* __restrict__ ei,
                                   const float* __restrict__ Q,
                                   const float* __restrict__ KP,
                                   const float* __restrict__ p_rel,
                                   float* __restrict__ alpha,
                                   int* __restrict__ m)
{
    int gid = blockIdx.x * blockDim.x + threadIdx.x;
    if (gid >= E_EDGES * H) return;
    const int e = gid >> 2, h = gid & 3;
    const int src = ei[e], dst = ei[E_EDGES + e];
    const v4f* q = (const v4f*)(Q + (size_t)dst * D + h * DH);
    const v4f* k = (const v4f*)(KP + (size_t)src * D + h * DH);
    float acc = 0.0f;
#pragma unroll
    for (int i = 0; i < 8; ++i) {
        v4f qa = q[i], ka = k[i];
        acc += qa.x * ka.x + qa.y * ka.y + qa.z * ka.z + qa.w * ka.w;
    }
    const float al = acc * p_rel[h] * INV_SQRT_DH;
    alpha[(size_t)e * H + h] = al;
    atomicMax(&m[(size_t)dst * H + h], ordf(al));
}

__global__ void edge_exp_kernel(const int* __restrict__ ei,
                                const int* __restrict__ m,
                                float* __restrict__ alpha,
                                float* __restrict__ s)
{
    int gid = blockIdx.x * blockDim.x + threadIdx.x;
    if (gid >= E_EDGES * H) return;
    const int e = gid >> 2, h = gid & 3;
    const int dst = ei[E_EDGES + e];
    const float mv = unordf(m[(size_t)dst * H + h]);
    const float ex = expf(alpha[gid] - mv);
    alpha[gid] = ex;
    atomicAdd(&s[(size_t)dst * H + h], ex);
}

__global__ void edge_scatter_kernel(const int* __restrict__ ei,
                                    const float* __restrict__ VP,
                                    const float* __restrict__ ex,
                                    const float* __restrict__ s,
                                    float* __restrict__ agg)
{
    long gid = (long)blockIdx.x * blockDim.x + threadIdx.x;
    if (gid >= (long)E_EDGES * D) return;
    const int e = (int)(gid >> 7);
    const int t = (int)(gid & 127);
    const int h = t >> 5;
    const int src = ei[e], dst = ei[E_EDGES + e];
    const float w = ex[(size_t)e * H + h] / (s[(size_t)dst * H + h] + 1e-16f);
    atomicAdd(&agg[(size_t)dst * D + t], VP[(size_t)src * D + t] * w);
}

__global__ void gelu_mean_kernel(float* __restrict__ agg, float invCnt, int n) {
    int gid = blockIdx.x * blockDim.x + threadIdx.x;
    if (gid < n) {
        const float v = agg[gid] * invCnt;
        agg[gid] = 0.5f * v * (1.0f + erff(v * 0.70710678118654752f));
    }
}

__global__ void copy_out_kernel(const float* __restrict__ zu,
                                const float* __restrict__ zi,
                                float* __restrict__ out)
{
    int gid = blockIdx.x * blockDim.x + threadIdx.x;
    const int n = BATCH * D;
    if (gid < n) { out[gid] = zu[gid]; out[n + gid] = zi[gid]; }
}

// ---------------------------------------------------------------------------
extern "C" void kernel_launch(void* const* d_in, const int* in_sizes, int n_in,
                              void* d_out, int out_size, void* d_ws, size_t ws_size,
                              hipStream_t stream)
{
    (void)in_sizes; (void)n_in; (void)out_size; (void)ws_size;

    const float* x_user = (const float*)d_in[0];
    const float* x_item = (const float*)d_in[1];
    const int* ei_ui = (const int*)d_in[2];
    const int* ei_iu = (const int*)d_in[3];
    const int* ei_uu = (const int*)d_in[4];

    const size_t NB = (size_t)N_NODES * D;
    float* w = (float*)d_ws;
    float* zu = w;     w += NB;
    float* zi = w;     w += NB;
    float* ku = w;     w += NB;
    float* qu = w;     w += NB;
    float* vu = w;     w += NB;
    float* ki = w;     w += NB;
    float* qi = w;     w += NB;
    float* vi = w;     w += NB;
    float* kp_ui = w;  w += NB;
    float* vp_ui = w;  w += NB;
    float* kp_iu = w;  w += NB;
    float* vp_iu = w;  w += NB;
    float* kp_uu = w;  w += NB;
    float* vp_uu = w;  w += NB;
    float* aggu = w;   w += NB;
    float* aggi = w;   w += NB;
    float* alphaB = w; w += (size_t)E_EDGES * H;
    int*   mB = (int*)w; w += (size_t)N_NODES * H;
    float* sB = w;     w += (size_t)N_NODES * H;

    const int MT = N_NODES / 16;                  // 6250 row tiles
    dim3 blk(256);
    dim3 gGemm((MT + 7) / 8);
    dim3 gRel((MT * H + 7) / 8);
    dim3 gEH((E_EDGES * H + 255) / 256);
    dim3 gED((unsigned)(((long)E_EDGES * D + 255) / 256));
    dim3 gNB((unsigned)((NB + 255) / 256));
    dim3 gNH((N_NODES * H + 255) / 256);
    dim3 gOut((BATCH * D + 255) / 256);

    for (int L = 0; L < 2; ++L) {
        const float* xu = (L == 0) ? x_user : zu;
        const float* xi = (L == 0) ? x_item : zi;
        const int base = 5 + L * 27;    // jax dict-sorted leaf order per layer
        // edge params: 'iu','ui','uu' x ('a_rel','m_rel','p_rel')
        const float* a_iu = (const float*)d_in[base + 0];
        const float* r_iu = (const float*)d_in[base + 1];
        const float* p_iu = (const float*)d_in[base + 2];
        const float* a_ui = (const float*)d_in[base + 3];
        const float* r_ui = (const float*)d_in[base + 4];
        const float* p_ui = (const float*)d_in[base + 5];
        const float* a_uu = (const float*)d_in[base + 6];
        const float* r_uu = (const float*)d_in[base + 7];
        const float* p_uu = (const float*)d_in[base + 8];
        // node params: 'item' then 'user' x ('Wa','Wk','Wq','Wv','ba','bk','bq','bv','skip')
        const float* Wa_i = (const float*)d_in[base + 9];
        const float* Wk_i = (const float*)d_in[base + 10];
        const float* Wq_i = (const float*)d_in[base + 11];
        const float* Wv_i = (const float*)d_in[base + 12];
        const float* ba_i = (const float*)d_in[base + 13];
        const float* bk_i = (const float*)d_in[base + 14];
        const float* bq_i = (const float*)d_in[base + 15];
        const float* bv_i = (const float*)d_in[base + 16];
        const float* sk_i = (const float*)d_in[base + 17];
        const float* Wa_u = (const float*)d_in[base + 18];
        const float* Wk_u = (const float*)d_in[base + 19];
        const float* Wq_u = (const float*)d_in[base + 20];
        const float* Wv_u = (const float*)d_in[base + 21];
        const float* ba_u = (const float*)d_in[base + 22];
        const float* bk_u = (const float*)d_in[base + 23];
        const float* bq_u = (const float*)d_in[base + 24];
        const float* bv_u = (const float*)d_in[base + 25];
        const float* sk_u = (const float*)d_in[base + 26];

        // --- k/q/v projections (WMMA) ---
        gemm128_kernel<<<gGemm, blk, 0, stream>>>(xu, Wk_u, bk_u, ku, N_NODES, 0, nullptr, nullptr);
        gemm128_kernel<<<gGemm, blk, 0, stream>>>(xu, Wq_u, bq_u, qu, N_NODES, 0, nullptr, nullptr);
        gemm128_kernel<<<gGemm, blk, 0, stream>>>(xu, Wv_u, bv_u, vu, N_NODES, 0, nullptr, nullptr);
        gemm128_kernel<<<gGemm, blk, 0, stream>>>(xi, Wk_i, bk_i, ki, N_NODES, 0, nullptr, nullptr);
        gemm128_kernel<<<gGemm, blk, 0, stream>>>(xi, Wq_i, bq_i, qi, N_NODES, 0, nullptr, nullptr);
        gemm128_kernel<<<gGemm, blk, 0, stream>>>(xi, Wv_i, bv_i, vi, N_NODES, 0, nullptr, nullptr);

        // --- per-head relation transforms (WMMA) ---
        relgemm_kernel<<<gRel, blk, 0, stream>>>(ku, a_ui, kp_ui, N_NODES);
        relgemm_kernel<<<gRel, blk, 0, stream>>>(vu, r_ui, vp_ui, N_NODES);
        relgemm_kernel<<<gRel, blk, 0, stream>>>(ki, a_iu, kp_iu, N_NODES);
        relgemm_kernel<<<gRel, blk, 0, stream>>>(vi, r_iu, vp_iu, N_NODES);
        relgemm_kernel<<<gRel, blk, 0, stream>>>(ku, a_uu, kp_uu, N_NODES);
        relgemm_kernel<<<gRel, blk, 0, stream>>>(vu, r_uu, vp_uu, N_NODES);

        // --- edge aggregation ---
        zero_kernel<<<gNB, blk, 0, stream>>>(aggu, (int)NB);
        zero_kernel<<<gNB, blk, 0, stream>>>(aggi, (int)NB);

        // ui: user -> item (attends with qi, aggregates into aggi)
        init_seg_kernel<<<gNH, blk, 0, stream>>>(mB, sB, N_NODES * H);
        edge_logits_kernel<<<gEH, blk, 0, stream>>>(ei_ui, qi, kp_ui, p_ui, alphaB, mB);
        edge_exp_kernel<<<gEH, blk, 0, stream>>>(ei_ui, mB, alphaB, sB);
        edge_scatter_kernel<<<gED, blk, 0, stream>>>(ei_ui, vp_ui, alphaB, sB, aggi);
        // iu: item -> user
        init_seg_kernel<<<gNH, blk, 0, stream>>>(mB, sB, N_NODES * H);
        edge_logits_kernel<<<gEH, blk, 0, stream>>>(ei_iu, qu, kp_iu, p_iu, alphaB, mB);
        edge_exp_kernel<<<gEH, blk, 0, stream>>>(ei_iu, mB, alphaB, sB);
        edge_scatter_kernel<<<gED, blk, 0, stream>>>(ei_iu, vp_iu, alphaB, sB, aggu);
        // uu: user -> user
        init_seg_kernel<<<gNH, blk, 0, stream>>>(mB, sB, N_NODES * H);
        edge_logits_kernel<<<gEH, blk, 0, stream>>>(ei_uu, qu, kp_uu, p_uu, alphaB, mB);
        edge_exp_kernel<<<gEH, blk, 0, stream>>>(ei_uu, mB, alphaB, sB);
        edge_scatter_kernel<<<gED, blk, 0, stream>>>(ei_uu, vp_uu, alphaB, sB, aggu);

        // --- mean across edge types + exact GELU ---
        gelu_mean_kernel<<<gNB, blk, 0, stream>>>(aggu, 0.5f, (int)NB);   // user: iu + uu
        gelu_mean_kernel<<<gNB, blk, 0, stream>>>(aggi, 1.0f, (int)NB);   // item: ui

        // --- output projection + sigmoid-skip residual (WMMA, fused epilogue) ---
        gemm128_kernel<<<gGemm, blk, 0, stream>>>(aggu, Wa_u, ba_u, zu, N_NODES, 1, sk_u, xu);
        gemm128_kernel<<<gGemm, blk, 0, stream>>>(aggi, Wa_i, ba_i, zi, N_NODES, 1, sk_i, xi);
    }

    copy_out_kernel<<<gOut, blk, 0, stream>>>(zu, zi, (float*)d_out);
}